// VGAE_38096359915635
// MI455X (gfx1250) — compile-verified
//
#include <hip/hip_runtime.h>
#include <hip/hip_bf16.h>

// ---------------------------------------------------------------------------
// VGAE forward pass on MI455X (gfx1250), fp32 end-to-end using
// V_WMMA_F32_16X16X4_F32 with register-blocked tiles:
//   - z@z^T : 32x64 output block per wave (8 accumulators, 8 WMMAs / K-step)
//   - GEMMs : 16x64 output block per wave (A reused across 4 N-tiles)
// ---------------------------------------------------------------------------

typedef __attribute__((ext_vector_type(2))) float v2f;
typedef __attribute__((ext_vector_type(8))) float v8f;

static constexpr int Nn   = 8192;
static constexpr int En   = 262144;
static constexpr int F_IN = 512;
static constexpr int H1   = 256;
static constexpr int H2   = 128;

#define WMMA_F32(a, b, c) \
  __builtin_amdgcn_wmma_f32_16x16x4_f32(false, (a), false, (b), (short)0, (c), false, false)

// ---------------------------------------------------------------------------
// fp32 WMMA GEMM, 16x64 block per wave: C[M,N] = A[M,K] @ B[K,N], row-major.
// Requires M%16==0, N%64==0, K%4==0.
//
// ISA layouts (CDNA5 ISA §7.12.2):
//   A 16x4  : lane L, VGPR v  ->  M = L%16, K = v + 2*(L/16)
//   B 4x16  : lane L, VGPR v  ->  K = v + 2*(L/16), N = L%16
//   C 16x16 : lane L, VGPR v  ->  M = v + 8*(L/16), N = L%16
// ---------------------------------------------------------------------------
__global__ __launch_bounds__(256) void wmma_gemm_f32_n4(
    const float* __restrict__ A, const float* __restrict__ B,
    float* __restrict__ C, int M, int N, int K) {
  const int wave = blockIdx.x * (blockDim.x >> 5) + (threadIdx.x >> 5);
  const int lane = threadIdx.x & 31;
  const int tn   = N >> 6;               // 64-wide N blocks
  const int mt   = wave / tn;
  const int nb   = wave % tn;
  if (mt >= (M >> 4)) return;            // wave-uniform: EXEC stays all-1s

  const int half = lane >> 4;
  const int lm   = lane & 15;

  const float* arow  = A + (size_t)(mt * 16 + lm) * K;
  const float* bbase = B + (size_t)(nb * 64 + lm);

  v8f acc0 = {}, acc1 = {}, acc2 = {}, acc3 = {};
  for (int kk = 0; kk < K; kk += 4) {
    const int k0 = kk + 2 * half;
    v2f a, b0, b1, b2, b3;
    a.x = arow[k0];
    a.y = arow[k0 + 1];
    const float* br0 = bbase + (size_t)k0 * N;
    const float* br1 = bbase + (size_t)(k0 + 1) * N;
    b0.x = br0[0];  b0.y = br1[0];
    b1.x = br0[16]; b1.y = br1[16];
    b2.x = br0[32]; b2.y = br1[32];
    b3.x = br0[48]; b3.y = br1[48];
    acc0 = WMMA_F32(a, b0, acc0);
    acc1 = WMMA_F32(a, b1, acc1);
    acc2 = WMMA_F32(a, b2, acc2);
    acc3 = WMMA_F32(a, b3, acc3);
  }

  float* crow = C + (size_t)(mt * 16 + 8 * half) * N + nb * 64 + lm;
#pragma unroll
  for (int v = 0; v < 8; ++v) {
    float* cr = crow + (size_t)v * N;
    cr[0]  = acc0[v];
    cr[16] = acc1[v];
    cr[32] = acc2[v];
    cr[48] = acc3[v];
  }
}

// ---------------------------------------------------------------------------
// adj_rec = Z @ Z^T : C[Nr,Nr], Z[Nr,K] row-major, 32x64 block per wave.
// B[k,n] = Z[n,k], so B tiles are read with the A-style (row-of-Z) indexing.
// Requires Nr%64==0, K%4==0.
// ---------------------------------------------------------------------------
__global__ __launch_bounds__(256) void wmma_zzT_f32_blk(
    const float* __restrict__ Z, float* __restrict__ C, int Nr, int K) {
  const int wave = blockIdx.x * (blockDim.x >> 5) + (threadIdx.x >> 5);
  const int lane = threadIdx.x & 31;
  const int tn   = Nr >> 6;              // 64-wide N blocks
  const int mb   = wave / tn;            // 32-tall M blocks
  const int nb   = wave % tn;
  if (mb >= (Nr >> 5)) return;           // wave-uniform

  const int half  = lane >> 4;
  const int lm    = lane & 15;
  const int mbase = mb * 32;
  const int nbase = nb * 64;

  const float* a0 = Z + (size_t)(mbase + lm) * K;
  const float* a1 = a0 + (size_t)16 * K;
  const float* b0 = Z + (size_t)(nbase + lm) * K;
  const float* b1 = b0 + (size_t)16 * K;
  const float* b2 = b0 + (size_t)32 * K;
  const float* b3 = b0 + (size_t)48 * K;

  v8f c00 = {}, c01 = {}, c02 = {}, c03 = {};
  v8f c10 = {}, c11 = {}, c12 = {}, c13 = {};
  for (int kk = 0; kk < K; kk += 4) {
    const int k0 = kk + 2 * half;
    v2f av0, av1, bv0, bv1, bv2, bv3;
    av0.x = a0[k0]; av0.y = a0[k0 + 1];
    av1.x = a1[k0]; av1.y = a1[k0 + 1];
    bv0.x = b0[k0]; bv0.y = b0[k0 + 1];
    bv1.x = b1[k0]; bv1.y = b1[k0 + 1];
    bv2.x = b2[k0]; bv2.y = b2[k0 + 1];
    bv3.x = b3[k0]; bv3.y = b3[k0 + 1];
    c00 = WMMA_F32(av0, bv0, c00);
    c01 = WMMA_F32(av0, bv1, c01);
    c02 = WMMA_F32(av0, bv2, c02);
    c03 = WMMA_F32(av0, bv3, c03);
    c10 = WMMA_F32(av1, bv0, c10);
    c11 = WMMA_F32(av1, bv1, c11);
    c12 = WMMA_F32(av1, bv2, c12);
    c13 = WMMA_F32(av1, bv3, c13);
  }

  float* crow0 = C + (size_t)(mbase + 8 * half) * Nr + nbase + lm;
  float* crow1 = crow0 + (size_t)16 * Nr;
#pragma unroll
  for (int v = 0; v < 8; ++v) {
    float* r0 = crow0 + (size_t)v * Nr;
    float* r1 = crow1 + (size_t)v * Nr;
    r0[0]  = c00[v];
    r0[16] = c01[v];
    r0[32] = c02[v];
    r0[48] = c03[v];
    r1[0]  = c10[v];
    r1[16] = c11[v];
    r1[32] = c12[v];
    r1[48] = c13[v];
  }
}

// ---------------------------------------------------------------------------
// SpMM (segment-sum over edges): out[row[e]] += edge_w[e] * H[col[e]]
// One block per edge; coalesced gather + fp32 global (L2) atomics.
// ---------------------------------------------------------------------------
__global__ __launch_bounds__(256) void spmm_d256(
    const int* __restrict__ row, const int* __restrict__ col,
    const float* __restrict__ ew, const float* __restrict__ H,
    float* __restrict__ out) {
  const int e = blockIdx.x;
  const int d = threadIdx.x;           // 0..255
  const int r = row[e];
  const int c = col[e];
  const float w = ew[e];
  atomicAdd(&out[(size_t)r * H1 + d], w * H[(size_t)c * H1 + d]);
}

// Fused mu/logvar SpMM (D = 128 each; 256 threads split 128/128).
__global__ __launch_bounds__(256) void spmm_mu_lv(
    const int* __restrict__ row, const int* __restrict__ col,
    const float* __restrict__ ew,
    const float* __restrict__ Pmu, const float* __restrict__ Plv,
    float* __restrict__ mu, float* __restrict__ lv) {
  const int e = blockIdx.x;
  const int t = threadIdx.x;
  const int r = row[e];
  const int c = col[e];
  const float w = ew[e];
  if (t < H2) {
    atomicAdd(&mu[(size_t)r * H2 + t], w * Pmu[(size_t)c * H2 + t]);
  } else {
    const int d = t - H2;
    atomicAdd(&lv[(size_t)r * H2 + d], w * Plv[(size_t)c * H2 + d]);
  }
}

__global__ __launch_bounds__(256) void relu_k(
    const float* __restrict__ in, float* __restrict__ out, int n) {
  int i = blockIdx.x * blockDim.x + threadIdx.x;
  if (i < n) out[i] = fmaxf(in[i], 0.0f);
}

// ---------------------------------------------------------------------------
extern "C" void kernel_launch(void* const* d_in, const int* in_sizes, int n_in,
                              void* d_out, int out_size, void* d_ws, size_t ws_size,
                              hipStream_t stream) {
  (void)in_sizes; (void)n_in; (void)out_size; (void)d_ws; (void)ws_size;

  const float* x   = (const float*)d_in[0];   // [N, F_IN]
  const float* W1  = (const float*)d_in[1];   // [F_IN, H1]
  const float* W2  = (const float*)d_in[2];   // [H1, H2]
  const float* W3  = (const float*)d_in[3];   // [H1, H2]
  const float* ew  = (const float*)d_in[4];   // [E]
  const int*   row = (const int*)d_in[5];     // [E]
  const int*   col = (const int*)d_in[6];     // [E]

  float* out = (float*)d_out;
  // Output layout (flat, return order): h1, adj_rec, mu, logvar, z
  float* h1  = out;                                   // N*H1
  float* adj = h1  + (size_t)Nn * H1;                 // N*N
  float* mu  = adj + (size_t)Nn * Nn;                 // N*H2
  float* lv  = mu  + (size_t)Nn * H2;                 // N*H2
  float* z   = lv  + (size_t)Nn * H2;                 // N*H2

  // Scratch carved from the adj_rec region (268 MB, written last by z@z^T,
  // which fully overwrites it). ~24 MB needed; no dependence on ws_size.
  float* h0     = adj;                                // N*H1  (x@W1, pre-spmm)
  float* s1     = h0     + (size_t)Nn * H1;           // N*H1  (spmm accum)
  float* mu_pre = s1     + (size_t)Nn * H1;           // N*H2
  float* lv_pre = mu_pre + (size_t)Nn * H2;           // N*H2

  // Zero the atomic-accumulate targets (s1 in scratch; mu+lv are contiguous).
  hipMemsetAsync(s1, 0, (size_t)Nn * H1 * sizeof(float), stream);
  hipMemsetAsync(mu, 0, (size_t)2 * Nn * H2 * sizeof(float), stream);

  // 1) h0 = x @ W1            [8192x512]x[512x256]
  {
    int waves = (Nn / 16) * (H1 / 64);                // 2048 waves, 8/block
    wmma_gemm_f32_n4<<<waves / 8, 256, 0, stream>>>(x, W1, h0, Nn, H1, F_IN);
  }
  // 2) s1 = spmm(h0); h1 = relu(s1)
  spmm_d256<<<En, H1, 0, stream>>>(row, col, ew, h0, s1);
  relu_k<<<(Nn * H1) / 256, 256, 0, stream>>>(s1, h1, Nn * H1);

  // 3) mu_pre = h1 @ W2 ; lv_pre = h1 @ W3    [8192x256]x[256x128]
  {
    int waves = (Nn / 16) * (H2 / 64);                // 1024 waves
    wmma_gemm_f32_n4<<<waves / 8, 256, 0, stream>>>(h1, W2, mu_pre, Nn, H2, H1);
    wmma_gemm_f32_n4<<<waves / 8, 256, 0, stream>>>(h1, W3, lv_pre, Nn, H2, H1);
  }
  // 4) mu = spmm(mu_pre); logvar = spmm(lv_pre)   (fused)
  spmm_mu_lv<<<En, 2 * H2, 0, stream>>>(row, col, ew, mu_pre, lv_pre, mu, lv);

  // 5) z = mu (eval-mode reparameterize)
  hipMemcpyAsync(z, mu, (size_t)Nn * H2 * sizeof(float),
                 hipMemcpyDeviceToDevice, stream);

  // 6) adj_rec = z @ z^T      [8192x128]x[128x8192] — overwrites scratch
  {
    int waves = (Nn / 32) * (Nn / 64);                // 32768 waves
    wmma_zzT_f32_blk<<<waves / 8, 256, 0, stream>>>(z, adj, Nn, H2);
  }
}